// Sparse4DHead1st_70248485094103
// MI455X (gfx1250) — compile-verified
//
#include <hip/hip_runtime.h>

// Problem constants (from reference)
#define NANCH   900
#define APAD    912          // 57 * 16, padded anchor count for WMMA M tiles
#define NCAMS   6
#define CCH     256
#define NPTS    13
#define NLVL    4
#define NGRP    8
#define OUTW    512          // concat(out[256], instance_feature[256])

typedef __attribute__((ext_vector_type(2))) float v2f;
typedef __attribute__((ext_vector_type(8))) float v8f;

// ---------------------------------------------------------------------------
// Phase 1: deformable aggregation -> acc[APAD][256] in workspace,
//          plus instance_feature copy into the concat half of d_out.
// One 64-thread sub-group per anchor; each thread owns 4 channels (float4).
// ---------------------------------------------------------------------------
__global__ __launch_bounds__(256) void daf_agg_kernel(
    const float* __restrict__ fm0, const float* __restrict__ fm1,
    const float* __restrict__ fm2, const float* __restrict__ fm3,
    const float* __restrict__ pts, const float* __restrict__ wts,
    const float* __restrict__ inst,
    float* __restrict__ acc, float* __restrict__ out)
{
    const int t  = threadIdx.x;
    const int a  = blockIdx.x * 4 + (t >> 6);   // anchor (0..911)
    const int ts = t & 63;                      // thread within anchor group
    const int c0 = ts * 4;                      // channel base (float4)
    const int g  = c0 >> 5;                     // weight group (rep = 32)

    float4 acc4 = make_float4(0.f, 0.f, 0.f, 0.f);

    if (a < NANCH) {
        const float* fms[NLVL] = {fm0, fm1, fm2, fm3};
        const int Hs[NLVL] = {64, 32, 16, 8};
        const int Ws[NLVL] = {176, 88, 44, 22};

        for (int p = 0; p < NPTS; ++p) {
            for (int cam = 0; cam < NCAMS; ++cam) {
                const int base = (a * NPTS + p) * NCAMS + cam;
                const float px = pts[base * 2 + 0];
                const float py = pts[base * 2 + 1];
                const float* wbase = wts + (size_t)base * (NLVL * NGRP) + g;
#pragma unroll
                for (int l = 0; l < NLVL; ++l) {
                    const int H = Hs[l], W = Ws[l];
                    const float x  = px * (float)W - 0.5f;
                    const float y  = py * (float)H - 0.5f;
                    const float xf = floorf(x);
                    const float yf = floorf(y);
                    const float wx1 = x - xf, wy1 = y - yf;
                    const float wx0 = 1.f - wx1, wy0 = 1.f - wy1;
                    const int x0 = (int)xf;
                    const int y0 = (int)yf;
                    const float w = wbase[l * NGRP];
                    const float* f = fms[l] + ((size_t)cam * H * W) * CCH + c0;
#pragma unroll
                    for (int dy = 0; dy < 2; ++dy) {
#pragma unroll
                        for (int dx = 0; dx < 2; ++dx) {
                            const int xi = x0 + dx;
                            const int yi = y0 + dy;
                            if (xi >= 0 && xi < W && yi >= 0 && yi < H) {
                                const float cw =
                                    (dx ? wx1 : wx0) * (dy ? wy1 : wy0) * w;
                                const float4 v = *(const float4*)(
                                    f + ((size_t)yi * W + xi) * CCH);
                                acc4.x += v.x * cw;
                                acc4.y += v.y * cw;
                                acc4.z += v.z * cw;
                                acc4.w += v.w * cw;
                            }
                        }
                    }
                }
            }
        }
        // concat residual: instance_feature -> out[:, 256:512]
        const float4 iv = *(const float4*)(inst + (size_t)a * CCH + c0);
        *(float4*)(out + (size_t)a * OUTW + CCH + c0) = iv;
    }
    // pad rows (a in [900,912)) write zeros so the GEMM loads are unguarded
    *(float4*)(acc + (size_t)a * CCH + c0) = acc4;
}

// ---------------------------------------------------------------------------
// Phase 2: out[:, 0:256] = acc @ w_proj + b_proj  via v_wmma_f32_16x16x4_f32.
// Block = 256 threads = 8 waves. Block covers one 16-row M tile x 128 cols.
// A tile (16x256 f32 = 16 KB) staged in LDS, shared by all 8 waves.
//
// Fragment layouts (CDNA5 ISA 7.12.2, wave32):
//   A 16x4 f32 : lanes 0-15 -> M=lane, {v0,v1}={K+0,K+1};
//                lanes16-31 -> M=lane-16, {v0,v1}={K+2,K+3}
//   B 4x16 f32 : lanes 0-15 -> N=lane, {v0,v1}={K+0,K+1};
//                lanes16-31 -> N=lane-16, {v0,v1}={K+2,K+3}
//   C 16x16 f32: vgpr r: lanes 0-15 -> (M=r, N=lane);
//                        lanes16-31 -> (M=r+8, N=lane-16)
// ---------------------------------------------------------------------------
__global__ __launch_bounds__(256) void proj_wmma_kernel(
    const float* __restrict__ acc,
    const float* __restrict__ wproj,
    const float* __restrict__ bias,
    float* __restrict__ out,
    float* __restrict__ dump)
{
    __shared__ float As[16 * CCH];   // 16 KB

    const int bM = blockIdx.x >> 1;          // 0..56
    const int bN = blockIdx.x & 1;           // 0..1
    const int m0 = bM * 16;
    const int t  = threadIdx.x;

    // cooperative A-tile stage: 4096 floats = 1024 float4, 256 thr x 4
    {
        const float4* src = (const float4*)(acc + (size_t)m0 * CCH);
        float4* dst = (float4*)As;
#pragma unroll
        for (int i = 0; i < 4; ++i)
            dst[t + 256 * i] = src[t + 256 * i];
    }
    __syncthreads();

    const int wave  = t >> 5;                // 0..7 -> N sub-tile
    const int lane  = t & 31;
    const int lrow  = lane & 15;             // M (for A) / N (for B,C) in tile
    const int khalf = (lane >> 4) << 1;      // 0 or 2
    const int n0    = bN * 128 + wave * 16;

    const float* bcol = wproj + n0 + lrow;   // column base in w_proj[K][N]

    v8f c = {};
#pragma unroll 4
    for (int k = 0; k < CCH; k += 4) {
        // A fragment: 2 adjacent f32 from LDS (8B aligned: k+khalf is even)
        const v2f afrag = *(const v2f*)(As + lrow * CCH + k + khalf);
        // B fragment: 2 f32 from w_proj, stride 256 apart (same column)
        v2f bfrag;
        bfrag.x = bcol[(size_t)(k + khalf + 0) * CCH];
        bfrag.y = bcol[(size_t)(k + khalf + 1) * CCH];
        c = __builtin_amdgcn_wmma_f32_16x16x4_f32(
            /*neg_a=*/false, afrag, /*neg_b=*/false, bfrag,
            /*c_mod=*/(short)0, c, /*reuse_a=*/false, /*reuse_b=*/false);
    }

    const float bv = bias[n0 + lrow];
    const int rowoff = (lane >> 4) * 8;
#pragma unroll
    for (int r = 0; r < 8; ++r) {
        const int m = m0 + rowoff + r;
        const float val = c[r] + bv;
        // tail rows (900..911) go to a dump slot: keeps EXEC uniform
        float* p = (m < NANCH) ? (out + (size_t)m * OUTW + n0 + lrow) : dump;
        *p = val;
    }
}

extern "C" void kernel_launch(void* const* d_in, const int* in_sizes, int n_in,
                              void* d_out, int out_size, void* d_ws, size_t ws_size,
                              hipStream_t stream)
{
    const float* fm0   = (const float*)d_in[0];
    const float* fm1   = (const float*)d_in[1];
    const float* fm2   = (const float*)d_in[2];
    const float* fm3   = (const float*)d_in[3];
    const float* pts   = (const float*)d_in[4];
    const float* wts   = (const float*)d_in[5];
    const float* inst  = (const float*)d_in[6];
    const float* wproj = (const float*)d_in[7];
    const float* bias  = (const float*)d_in[8];
    float* out = (float*)d_out;

    float* acc  = (float*)d_ws;                 // APAD*256 floats
    float* dump = acc + (size_t)APAD * CCH;     // scratch for tail-row stores

    // Phase 1: aggregation (+ concat copy). 228 blocks * 4 anchors = 912.
    daf_agg_kernel<<<APAD / 4, 256, 0, stream>>>(
        fm0, fm1, fm2, fm3, pts, wts, inst, acc, out);

    // Phase 2: projection GEMM. 57 M-tiles * 2 N-halves.
    proj_wmma_kernel<<<57 * 2, 256, 0, stream>>>(acc, wproj, bias, out, dump);
}